// HighOrder_88802743812152
// MI455X (gfx1250) — compile-verified
//
#include <hip/hip_runtime.h>

typedef __attribute__((ext_vector_type(2))) float v2f;
typedef __attribute__((ext_vector_type(4))) float v4f;
typedef __attribute__((ext_vector_type(8))) float v8f;

#define TILE 32
#define HALO 2
#define PAD  (TILE + 2*HALO)          // 36
#define CSTR (PAD*PAD)                // 1296
#define CIN  3
#define KTOT 75
#define KPAD 76
#define NOUT 25
#define WSTR 36                       // m-stride of transposed tap spill (16B mult)
#define IMH  512
#define IMW  512

// K-space permutation: kp 0..59 are "kw pairs" ((kw0,kw1),(kw2,kw3) of each of
// the 15 (c,kh) rows), kp 60..74 are the kw=4 leftovers, kp 75 is zero pad.
// Original k = c*25 + kh*5 + kw = 5*t + kw with t = c*5 + kh.
__host__ __device__ constexpr int korig(int kp) {
    return (kp < 60) ? (5 * ((kp / 2) / 2) + 2 * ((kp / 2) % 2) + (kp % 2))
         : (kp < 75) ? (5 * (kp - 60) + 4)
                     : 75;  // pad
}

// LDS offset of original patch element k relative to patch top-left (r*PAD+col).
__host__ __device__ constexpr int patchOff(int k) {
    return (k / 25) * CSTR + ((k % 25) / 5) * PAD + (k % 5);
}

__global__ __launch_bounds__(256)
void dynconv_fused(const float* __restrict__ x,
                   const float* __restrict__ cw,
                   const float* __restrict__ cb,
                   float* __restrict__ out) {
    __shared__ float sX[CIN * CSTR];              // input tile + halo (3x36x36)
    __shared__ float sB[KPAD * 32];               // permuted weights [kp][n]
    __shared__ float sBias[32];
    __shared__ __align__(16) float sWvT[8][NOUT][WSTR];  // per-wave taps [n][m]

    const int tid  = threadIdx.x;
    const int lane = tid & 31;
    const int wv   = tid >> 5;
    const int hi   = lane >> 4;                   // 0: lanes 0-15, 1: 16-31
    const int lo16 = lane & 15;

    const int tx0 = blockIdx.x * TILE;
    const int ty0 = blockIdx.y * TILE;
    const int b   = blockIdx.z;

    // ---- stage x tile (zero halo) into LDS ----
    for (int i = tid; i < CIN * CSTR; i += 256) {
        int c  = i / CSTR;
        int rr = (i - c * CSTR) / PAD;
        int qq = i % PAD;
        int gy = ty0 - HALO + rr;
        int gx = tx0 - HALO + qq;
        float v = 0.f;
        if ((unsigned)gy < IMH && (unsigned)gx < IMW)
            v = x[(((size_t)b * CIN + c) * IMH + gy) * IMW + gx];
        sX[i] = v;
    }
    // ---- permuted/padded weights: sB[kp][n] = cw[n*75 + korig(kp)] ----
    for (int i = tid; i < KPAD * 32; i += 256) {
        int kp = i >> 5;
        int n  = i & 31;
        sB[i] = (kp < KTOT && n < NOUT) ? cw[n * KTOT + korig(kp)] : 0.f;
    }
    if (tid < 32) sBias[tid] = (tid < NOUT) ? cb[tid] : 0.f;
    __syncthreads();

    // ---- preload B fragments (lanes 0-15: K=kb,kb+1 at N=lane;
    //      lanes 16-31: K=kb+2,kb+3 at N=lane-16) ----
    v2f Bfrag[19][2];
#pragma unroll
    for (int kk = 0; kk < 19; ++kk) {
        const int kb = 4 * kk + 2 * hi;
#pragma unroll
        for (int nt = 0; nt < 2; ++nt) {
            const int n = lo16 + 16 * nt;
            v2f bf;
            bf.x = sB[kb * 32 + n];
            bf.y = sB[(kb + 1) * 32 + n];
            Bfrag[kk][nt] = bf;
        }
    }

    // bias splat as initial C operand (C layout: N in lane, 8 M-rows per lane)
    const float bias0 = sBias[lo16];
    const float bias1 = sBias[lo16 + 16];         // zero-padded for n >= 25
    v8f biasC0, biasC1;
#pragma unroll
    for (int g = 0; g < 8; ++g) { biasC0[g] = bias0; biasC1[g] = bias1; }

    // per-lane A-fragment offsets (hi-half selected once, hoisted out of loop)
    int aoffP[15];
#pragma unroll
    for (int kk = 0; kk < 15; ++kk)
        aoffP[kk] = hi ? patchOff(korig(4 * kk + 2)) : patchOff(korig(4 * kk));
    int aoffS0[4], aoffS1[4];
#pragma unroll
    for (int t = 0; t < 4; ++t) {
        const int kk = 15 + t;
        aoffS0[t] = hi ? patchOff(korig(4 * kk + 2)) : patchOff(korig(4 * kk));
        aoffS1[t] = hi ? patchOff(korig(4 * kk + 3 < KTOT ? 4 * kk + 3
                                                          : 4 * kk + 2))
                       : patchOff(korig(4 * kk + 1));
    }

    // each wave owns 4 rows of the 32x32 tile; one chunk = one row (32 px).
    // Keep this loop ROLLED: unrolling 4 bodies of 76 WMMAs explodes register
    // pressure (remat movs) without adding useful ILP beyond the 4 chains.
#pragma unroll 1
    for (int rci = 0; rci < 4; ++rci) {
        const int r = wv * 4 + rci;
        const int rowBase = r * PAD + lo16;       // patch top-left, m-tile 0

        // ============ stage A: implicit GEMM via fp32 WMMA =================
        // One shared base pointer per K-step: m-tile 1 rides in the DS
        // immediate offsets (offset0:16 offset1:17).
        v8f Cacc[2][2];
        {   // kk = 0 seeds the chains with the bias splat
            const float* pA = &sX[rowBase + aoffP[0]];
            v2f a0; a0.x = pA[0];  a0.y = pA[1];
            v2f a1; a1.x = pA[16]; a1.y = pA[17];
            Cacc[0][0] = __builtin_amdgcn_wmma_f32_16x16x4_f32(
                false, a0, false, Bfrag[0][0], (short)0, biasC0, false, false);
            Cacc[0][1] = __builtin_amdgcn_wmma_f32_16x16x4_f32(
                false, a0, false, Bfrag[0][1], (short)0, biasC1, false, false);
            Cacc[1][0] = __builtin_amdgcn_wmma_f32_16x16x4_f32(
                false, a1, false, Bfrag[0][0], (short)0, biasC0, false, false);
            Cacc[1][1] = __builtin_amdgcn_wmma_f32_16x16x4_f32(
                false, a1, false, Bfrag[0][1], (short)0, biasC1, false, false);
        }
#pragma unroll
        for (int kk = 1; kk < 15; ++kk) {
            const float* pA = &sX[rowBase + aoffP[kk]];
            v2f a0; a0.x = pA[0];  a0.y = pA[1];
            v2f a1; a1.x = pA[16]; a1.y = pA[17];
            Cacc[0][0] = __builtin_amdgcn_wmma_f32_16x16x4_f32(
                false, a0, false, Bfrag[kk][0], (short)0, Cacc[0][0], false, false);
            Cacc[0][1] = __builtin_amdgcn_wmma_f32_16x16x4_f32(
                false, a0, false, Bfrag[kk][1], (short)0, Cacc[0][1], false, false);
            Cacc[1][0] = __builtin_amdgcn_wmma_f32_16x16x4_f32(
                false, a1, false, Bfrag[kk][0], (short)0, Cacc[1][0], false, false);
            Cacc[1][1] = __builtin_amdgcn_wmma_f32_16x16x4_f32(
                false, a1, false, Bfrag[kk][1], (short)0, Cacc[1][1], false, false);
        }
        // kk = 15..18: kw=4 leftovers (+1 zero pad at K=75, hi lanes of kk=18)
#pragma unroll
        for (int t = 0; t < 4; ++t) {
            const int kk = 15 + t;
            const float* p0 = &sX[rowBase + aoffS0[t]];
            const float* p1 = &sX[rowBase + aoffS1[t]];
            v2f a0, a1;
            a0.x = p0[0];  a1.x = p0[16];
            float y0 = p1[0], y1 = p1[16];
            if (t == 3) { y0 = hi ? 0.f : y0; y1 = hi ? 0.f : y1; }
            a0.y = y0; a1.y = y1;
            Cacc[0][0] = __builtin_amdgcn_wmma_f32_16x16x4_f32(
                false, a0, false, Bfrag[kk][0], (short)0, Cacc[0][0], false, false);
            Cacc[0][1] = __builtin_amdgcn_wmma_f32_16x16x4_f32(
                false, a0, false, Bfrag[kk][1], (short)0, Cacc[0][1], false, false);
            Cacc[1][0] = __builtin_amdgcn_wmma_f32_16x16x4_f32(
                false, a1, false, Bfrag[kk][0], (short)0, Cacc[1][0], false, false);
            Cacc[1][1] = __builtin_amdgcn_wmma_f32_16x16x4_f32(
                false, a1, false, Bfrag[kk][1], (short)0, Cacc[1][1], false, false);
        }

        // ---- spill taps transposed: sWvT[n][m], b128 stores from C regs ----
        // C layout: lane holds N = lane%16 (+16 for nt=1), M = (lane/16)*8 + g
        const int mB = hi * 8;
        const int n1ok = (lo16 + 16) < NOUT;
#pragma unroll
        for (int mt = 0; mt < 2; ++mt) {
            const int m = mt * 16 + mB;
            v4f lo0 = { Cacc[mt][0][0], Cacc[mt][0][1], Cacc[mt][0][2], Cacc[mt][0][3] };
            v4f hi0 = { Cacc[mt][0][4], Cacc[mt][0][5], Cacc[mt][0][6], Cacc[mt][0][7] };
            *(v4f*)&sWvT[wv][lo16][m]     = lo0;
            *(v4f*)&sWvT[wv][lo16][m + 4] = hi0;
            if (n1ok) {
                v4f lo1 = { Cacc[mt][1][0], Cacc[mt][1][1], Cacc[mt][1][2], Cacc[mt][1][3] };
                v4f hi1 = { Cacc[mt][1][4], Cacc[mt][1][5], Cacc[mt][1][6], Cacc[mt][1][7] };
                *(v4f*)&sWvT[wv][lo16 + 16][m]     = lo1;
                *(v4f*)&sWvT[wv][lo16 + 16][m + 4] = hi1;
            }
        }

        // ============ stage B: apply dynamic filter + residual =============
        // lane owns pixel (r, lane); per-channel base pointers keep every
        // patch offset within the DS immediate range (<=148 elements).
        const int pb = r * PAD + lane;
        const float* pP0 = &sX[0 * CSTR + pb];
        const float* pP1 = &sX[1 * CSTR + pb];
        const float* pP2 = &sX[2 * CSTR + pb];
        const float* pW  = &sWvT[wv][0][lane];
        float acc0 = pP0[2 * PAD + 2];
        float acc1 = pP1[2 * PAD + 2];
        float acc2 = pP2[2 * PAD + 2];
#pragma unroll
        for (int kh = 0; kh < 5; ++kh) {
#pragma unroll
            for (int kw = 0; kw < 5; ++kw) {
                const float wk = pW[(kh * 5 + kw) * WSTR];
                const int po = kh * PAD + kw;
                acc0 += wk * pP0[po];
                acc1 += wk * pP1[po];
                acc2 += wk * pP2[po];
            }
        }
        const int gy = ty0 + r;
        const int gx = tx0 + lane;
        const size_t o = (((size_t)b * CIN) * IMH + gy) * IMW + gx;
        out[o]                         = acc0;
        out[o + (size_t)IMH * IMW]     = acc1;
        out[o + 2 * (size_t)IMH * IMW] = acc2;
    }
}

extern "C" void kernel_launch(void* const* d_in, const int* in_sizes, int n_in,
                              void* d_out, int out_size, void* d_ws, size_t ws_size,
                              hipStream_t stream) {
    const float* x  = (const float*)d_in[0];
    const float* cw = (const float*)d_in[1];
    const float* cb = (const float*)d_in[2];
    float* out = (float*)d_out;

    const int B = in_sizes[0] / (CIN * IMH * IMW);   // 16
    dim3 grid(IMW / TILE, IMH / TILE, B);
    dynconv_fused<<<grid, 256, 0, stream>>>(x, cw, cb, out);
}